// SkeletonConv_58961311039803
// MI455X (gfx1250) — compile-verified
//
#include <hip/hip_runtime.h>
#include <stdint.h>

typedef float v2f  __attribute__((ext_vector_type(2)));
typedef float v8f  __attribute__((ext_vector_type(8)));
typedef unsigned int u32x4 __attribute__((ext_vector_type(4)));
typedef int   i32x4 __attribute__((ext_vector_type(4)));
typedef int   i32x8 __attribute__((ext_vector_type(8)));

#define B_    64
#define CIN   16
#define COUT  16
#define KTAPS 3
#define H_    256
#define W_    256
#define PITCH 264   // channel pitch in LDS (floats): 2*264 % 64 == 16 -> half-waves
                    // hit disjoint bank groups (no 2-way conflict on ds loads)

// One block (8 wave32) per (b,h) row.
//  Stage 1: TDM (tensor_load_to_lds) pulls the 3 tap rows (16ch x 256w fp32 each,
//           16 KB) into LDS with DMA-inserted pitch padding; OOB rows are
//           cooperatively zeroed. Tracked by TENSORcnt.
//  Stage 2: each wave computes 16x16 output tiles (16 C_out x 16 w-positions)
//           with V_WMMA_F32_16X16X4_F32; B fragments come from LDS.
__global__ __launch_bounds__(256) void skel_conv_wmma(
    const float* __restrict__ x,     // [B, CIN, H, W]
    const float* __restrict__ wgt,   // [COUT, CIN, KTAPS]
    const int*  __restrict__ dstep,  // scalar
    const int*  __restrict__ ddil,   // scalar
    float* __restrict__ out)         // [B, COUT, H, W]
{
    __shared__ float lds[KTAPS * CIN * PITCH];   // ~49.5 KB

    const int step = dstep[0];
    const int dil  = ddil[0];

    const int bh   = blockIdx.x;
    const int b    = bh / H_;
    const int h    = bh % H_;
    const int tid  = threadIdx.x;
    const int wave = tid >> 5;
    const int lane = tid & 31;
    const int n    = lane & 15;   // N index (w position) / M row for A
    const int hi   = lane >> 4;   // half-wave selector
    const int kkb  = hi * 2;      // K sub-index base within a 4-wide chunk

    const size_t plane = (size_t)H_ * W_;
    const float* xb = x + (size_t)b * CIN * plane;

    // ---------------- Stage 1: stage tap rows into LDS ----------------
#if __has_builtin(__builtin_amdgcn_tensor_load_to_lds)
    if (wave == 0) {
        // Tensor instructions ignore EXEC; issue from wave 0 only (uniform branch).
#pragma unroll
        for (int k = 0; k < KTAPS; ++k) {
            const int hh = h + (k - KTAPS / 2) * step;
            if (hh >= 0 && hh < H_) {
                const uint64_t gaddr =
                    (uint64_t)(uintptr_t)(xb + (size_t)hh * W_);
                const unsigned ldsAddr =
                    (unsigned)(uintptr_t)&lds[k * CIN * PITCH]; // low 32 bits = LDS byte offset

                // D# group 0: count=1 | lds_addr | global_addr[56:0] | type=2
                u32x4 g0;
                g0[0] = 1u;                               // count=1, user descriptor
                g0[1] = ldsAddr;                          // LDS byte address
                g0[2] = (unsigned)(gaddr & 0xffffffffu);  // global addr [31:0]
                g0[3] = (unsigned)((gaddr >> 32) & 0x01ffffffu) // [56:32]
                        | 0x80000000u;                    // type = 2 ("image")

                // D# group 1: data_size=4B, pad every 256 DW by 8 DW,
                // tensor_dim0=W (w, contiguous), tensor_dim1=CIN (c),
                // tile = full 256 x 16, dim0_stride = H*W elements.
                i32x8 g1;
                g1[0] = (int)((2u << 16)        // data_size = 4 bytes
                            | (1u << 20)        // pad_enable
                            | (7u << 22)        // pad_interval: 256 DWORDs
                            | (7u << 25));      // pad_amount:   8 DWORDs
                g1[1] = (int)((unsigned)W_ << 16);        // tensor_dim0[15:0]
                g1[2] = (int)((unsigned)CIN << 16);       // dim0 hi=0 | tensor_dim1[15:0]
                g1[3] = (int)((unsigned)W_ << 16);        // dim1 hi=0 | tile_dim0
                g1[4] = CIN;                              // tile_dim1 | tile_dim2=0
                g1[5] = (int)plane;                       // tensor_dim0_stride[31:0]
                g1[6] = 0;                                // stride hi | dim1_stride lo
                g1[7] = 0;

                const i32x4 gz4 = {0, 0, 0, 0};           // groups 2/3: 2D tensor
                const i32x8 gz8 = {0, 0, 0, 0, 0, 0, 0, 0};
                // 6-arg variant (clang-23 / therock-10.0 toolchain)
                __builtin_amdgcn_tensor_load_to_lds(g0, g1, gz4, gz4, gz8, /*cpol=*/0);
            }
        }
    }
#endif
    // Zero-fill slabs for out-of-range tap rows (and fallback copy path).
#pragma unroll
    for (int k = 0; k < KTAPS; ++k) {
        const int hh = h + (k - KTAPS / 2) * step;
        if (hh < 0 || hh >= H_) {
            for (int i = tid; i < CIN * PITCH; i += 256)
                lds[k * CIN * PITCH + i] = 0.0f;
        }
#if !__has_builtin(__builtin_amdgcn_tensor_load_to_lds)
        else {
            for (int i = tid; i < CIN * W_; i += 256) {
                const int c = i >> 8, w = i & (W_ - 1);
                lds[(k * CIN + c) * PITCH + w] = xb[(size_t)c * plane + (size_t)hh * W_ + w];
            }
        }
#endif
    }
#if __has_builtin(__builtin_amdgcn_s_wait_tensorcnt)
    __builtin_amdgcn_s_wait_tensorcnt(0);   // wave0: TDM landed; others: no-op
#endif
    __syncthreads();

    // ---------------- A matrices: weights resident in registers ----------------
    // A[k][q]: 16x4 fp32 = W[o=0..15, c=4q..4q+3, tap=k]
    // Layout: lane L -> M=L%16 ; VGPR0/1 = K {0,1} (lanes 0-15), {2,3} (lanes 16-31)
    v2f A[KTAPS][4];
#pragma unroll
    for (int k = 0; k < KTAPS; ++k) {
#pragma unroll
        for (int q = 0; q < 4; ++q) {
            const int c0 = 4 * q + kkb;
            A[k][q].x = wgt[(n * CIN + c0    ) * KTAPS + k];
            A[k][q].y = wgt[(n * CIN + c0 + 1) * KTAPS + k];
        }
    }

    // ---------------- Stage 2: WMMA over w-tiles ----------------
    for (int t = wave; t < W_ / 16; t += 8) {   // 16 tiles, 8 waves
        const int w0 = t * 16;
        v8f acc = {};   // 16x16 fp32 C/D tile

#pragma unroll
        for (int k = 0; k < KTAPS; ++k) {
            const int ww  = w0 + n + (k - KTAPS / 2) * dil;
            const bool wok = (ww >= 0) && (ww < W_);
            const int wwc = wok ? ww : 0;
            const float* lk = &lds[(k * CIN + kkb) * PITCH + wwc];

#pragma unroll
            for (int q = 0; q < 4; ++q) {
                // B: 4x16 fp32 tile from LDS (channels 4q..4q+3, positions w0..w0+15)
                v2f Bm;
                Bm.x = wok ? lk[(4 * q    ) * PITCH] : 0.0f;
                Bm.y = wok ? lk[(4 * q + 1) * PITCH] : 0.0f;
                acc = __builtin_amdgcn_wmma_f32_16x16x4_f32(
                        /*neg_a=*/false, A[k][q],
                        /*neg_b=*/false, Bm,
                        /*c_mod=*/(short)0, acc,
                        /*reuse_a=*/false, /*reuse_b=*/false);
            }
        }

        // Store: C/D VGPR r holds (co = r + 8*hi, position = w0 + n)
        float* ob = out + ((size_t)b * COUT + 8 * hi) * plane
                        + (size_t)h * W_ + w0 + n;
#pragma unroll
        for (int r = 0; r < 8; ++r)
            ob[(size_t)r * plane] = acc[r];
    }
}

extern "C" void kernel_launch(void* const* d_in, const int* in_sizes, int n_in,
                              void* d_out, int out_size, void* d_ws, size_t ws_size,
                              hipStream_t stream) {
    const float* x    = (const float*)d_in[0];
    const float* wgt  = (const float*)d_in[1];
    const int*   dil  = (const int*)d_in[2];
    const int*   step = (const int*)d_in[3];
    float*       out  = (float*)d_out;

    dim3 grid(B_ * H_);   // one block per (b, h) row
    dim3 block(256);      // 8 wave32
    skel_conv_wmma<<<grid, block, 0, stream>>>(x, wgt, step, dil, out);
}